// IntervalTimeEncoder_46651934769846
// MI455X (gfx1250) — compile-verified
//
#include <hip/hip_runtime.h>
#include <stdint.h>

// IntervalTimeEncoder for MI455X (gfx1250, wave32).
//
// out[b,l,e] = W[e, idx(b,l)] + bias[e],  idx = clamp((ts[l]-ts[l-1])/1000, 0, 100)
//
// Pure gather+bias: 256MB fp32 output stream, ~11us HBM roofline at 23.3TB/s.
// CDNA5 paths used:
//   - global_load_async_to_lds_b128 (ASYNCcnt) to stage W into LDS
//   - bias-fused transposed LDS table, conflict-free ds_load_b128 per row
//   - v_readlane bucket-id broadcast across the wave32
//   - non-temporal global_store_b128 (output > 192MB L2, don't thrash it)

#define B_DIM 128
#define L_DIM 4096
#define EMB   128
#define NBINS 101              // n_bins + 1
#define ROWS  (B_DIM * L_DIM)  // 524288

typedef __attribute__((ext_vector_type(4))) float v4f;

__global__ __launch_bounds__(256) void
time_encoder_kernel(const float* __restrict__ ts,    // [B, L]
                    const float* __restrict__ Wg,    // [EMB, NBINS] row-major
                    const float* __restrict__ bias,  // [EMB]
                    float* __restrict__ out)         // [B, L, EMB]
{
    // LDS: raw W staging (async-copied) + bias-fused transposed table
    __shared__ float stage[EMB * NBINS];        // 51712 B, W row-major [e][i]
    __shared__ v4f   tbl4[NBINS * (EMB / 4)];   // 51712 B, tbl[i][e] = W[e][i] + bias[e]
    float* tbl = (float*)tbl4;

    const int tid = threadIdx.x;

    // ---- Stage W into LDS with CDNA5 async global->LDS DMA (ASYNCcnt) ----
    // EMB*NBINS = 12928 floats = 3232 x b128 transfers; 256 threads -> 13 rounds.
    for (int j4 = tid; j4 < (EMB * NBINS) / 4; j4 += 256) {
        // Generic LDS pointer low 32 bits == wave-relative LDS byte offset.
        unsigned loff = (unsigned)(uintptr_t)(&stage[j4 * 4]);
        const float* gp = Wg + j4 * 4;
        asm volatile("global_load_async_to_lds_b128 %0, %1, off"
                     :: "v"(loff), "v"(gp) : "memory");
    }
    asm volatile("s_wait_asynccnt 0" ::: "memory");
    __syncthreads();

    // ---- Build bias-fused transposed table: tbl[i*128 + e] = stage[e*101 + i] + bias[e]
    // LDS read stride 101 floats (bank stride 37, odd) -> conflict-free.
    for (int j = tid; j < NBINS * EMB; j += 256) {
        int i = j >> 7;        // bucket id
        int e = j & (EMB - 1); // embedding element
        tbl[j] = stage[e * NBINS + i] + bias[e];
    }
    __syncthreads();

    // ---- Streaming gather: each wave handles 32 rows per outer step ----
    const int lane       = tid & 31;
    const int waveId     = blockIdx.x * (blockDim.x >> 5) + (tid >> 5);
    const int totalWaves = gridDim.x * (blockDim.x >> 5);

    v4f* __restrict__ out4 = (v4f*)out;

    for (long base = (long)waveId * 32; base < (long)ROWS;
         base += (long)totalWaves * 32) {
        // Coalesced per-lane bucket-id computation for 32 consecutive rows.
        int   r  = (int)base + lane;
        int   l  = r & (L_DIM - 1);
        float t1 = ts[r];
        float t0 = (l == 0) ? t1 : ts[r - 1];
        float pt = t1 - t0;
        int raw  = (int)(pt / 100000.0f * 100.0f);   // trunc toward zero, as astype(int32)
        int idx  = raw < 0 ? 0 : (raw > NBINS - 1 ? NBINS - 1 : raw);

        // Broadcast each row's id; 32 lanes emit one 512B row per step.
        #pragma unroll
        for (int rr = 0; rr < 32; ++rr) {
            int id = __builtin_amdgcn_readlane(idx, rr);        // scalar id
            v4f v  = tbl4[id * (EMB / 4) + lane];               // ds_load_b128, bank-clean
            __builtin_nontemporal_store(v, &out4[(base + rr) * (EMB / 4) + lane]);
        }
    }
}

extern "C" void kernel_launch(void* const* d_in, const int* in_sizes, int n_in,
                              void* d_out, int out_size, void* d_ws, size_t ws_size,
                              hipStream_t stream) {
    // d_in[0]: inputs int64 [B,L]  (unused by the reference output)
    // d_in[1]: timestamp float32 [B,L]
    // d_in[2]: W float32 [EMB, NBINS]
    // d_in[3]: b float32 [EMB]
    (void)in_sizes; (void)n_in; (void)d_ws; (void)ws_size; (void)out_size;

    const float* ts   = (const float*)d_in[1];
    const float* W    = (const float*)d_in[2];
    const float* bias = (const float*)d_in[3];
    float*       out  = (float*)d_out;

    // 1024 blocks x 8 waves x 32 rows = 262144 rows per sweep; 2 sweeps cover 524288.
    time_encoder_kernel<<<dim3(1024), dim3(256), 0, stream>>>(ts, W, bias, out);
}